// PitchesMultiheadAttention_15925738734088
// MI455X (gfx1250) — compile-verified
//
#include <hip/hip_runtime.h>

// MI455X / gfx1250 — wave32, WMMA f32 16x16x4 (exact fp32 matrix path).
typedef __attribute__((ext_vector_type(2))) float v2f;
typedef __attribute__((ext_vector_type(8))) float v8f;

#define SEQ    2048
#define BATCH  2
#define EMB    1024
#define HEADS  16
#define DHEAD  64
#define SCALE2 0.03125f  /* EMB^-0.5 : both q and k scaled by EMB^-0.25 */

__device__ __forceinline__ v8f wmma_f32(v2f a, v2f b, v8f c) {
  // D = A(16x4 f32) * B(4x16 f32) + C(16x16 f32)
  return __builtin_amdgcn_wmma_f32_16x16x4_f32(false, a, false, b, (short)0, c,
                                               false, false);
}

__device__ __forceinline__ float hmax16(float v) {
  v = fmaxf(v, __shfl_xor(v, 1, 32));
  v = fmaxf(v, __shfl_xor(v, 2, 32));
  v = fmaxf(v, __shfl_xor(v, 4, 32));
  v = fmaxf(v, __shfl_xor(v, 8, 32));
  return v;
}
__device__ __forceinline__ float hsum16(float v) {
  v += __shfl_xor(v, 1, 32);
  v += __shfl_xor(v, 2, 32);
  v += __shfl_xor(v, 4, 32);
  v += __shfl_xor(v, 8, 32);
  return v;
}

// ---------------------------------------------------------------------------
// Kernel 1: per-head projections.
//   q [b,h,s,o]  = X_h @ Wq[h]^T                       (unscaled, as reference)
//   kp[b,h,s,o]  = (X_h @ Wk[h]^T) * SCALE2 + rel_pos[o, s]   (folded K')
//   v [b,h,s,o]  =  X_h @ Wv[h]^T
// grid.x = B*H*(S/64), grid.y = proj in {0,1,2}; 4 waves, 16 rows each.
// ---------------------------------------------------------------------------
__global__ void __launch_bounds__(128)
qkv_kernel(const float* __restrict__ x, const float* __restrict__ Wq,
           const float* __restrict__ Wk, const float* __restrict__ Wv,
           const float* __restrict__ rel_pos,
           float* __restrict__ q, float* __restrict__ kp, float* __restrict__ v)
{
  const int proj = blockIdx.y;
  const int bid  = blockIdx.x;        // b*512 + h*32 + sblk
  const int sblk = bid & 31;
  const int h    = (bid >> 5) & 15;
  const int b    = bid >> 9;
  const int lane = threadIdx.x & 31;
  const int wave = threadIdx.x >> 5;
  const int g    = lane >> 4;         // half-wave selector
  const int ln   = lane & 15;

  const float* W    = (proj == 0) ? Wq : (proj == 1) ? Wk : Wv;
  float*       outp = (proj == 0) ? q  : (proj == 1) ? kp : v;

  const int srow = sblk * 64 + wave * 16;

  // A fragments: rows = s, K = input channel c. 16 chunks of K=4.
  v2f afrag[16];
  {
    const int s = srow + ln;
    const float* xrow = x + (size_t)s * (BATCH * EMB) + (size_t)b * EMB + h * DHEAD;
#pragma unroll
    for (int kb = 0; kb < 16; ++kb) {
      const int c0 = kb * 4 + 2 * g;
      afrag[kb].x = xrow[c0];
      afrag[kb].y = xrow[c0 + 1];
    }
  }

#pragma unroll
  for (int nt = 0; nt < 4; ++nt) {
    v8f acc;
#pragma unroll
    for (int r = 0; r < 8; ++r) acc[r] = 0.0f;

    const int o = nt * 16 + ln;
    const float* wrow = W + (size_t)h * DHEAD * DHEAD + (size_t)o * DHEAD;
#pragma unroll
    for (int kb = 0; kb < 16; ++kb) {
      const int c0 = kb * 4 + 2 * g;
      v2f bf;
      bf.x = wrow[c0];
      bf.y = wrow[c0 + 1];
      acc = wmma_f32(afrag[kb], bf, acc);
    }
#pragma unroll
    for (int r = 0; r < 8; ++r) {
      const int s = srow + r + 8 * g;
      float val = acc[r];
      if (proj == 1) val = val * SCALE2 + rel_pos[(size_t)o * SEQ + s];
      outp[((size_t)(b * HEADS + h) * SEQ + s) * DHEAD + o] = val;
    }
  }
}

// ---------------------------------------------------------------------------
// Kernel 2: causal flash attention per (b,h,64-query-rows).
//   scores = Q @ K'^T  (rel-bias + scale already folded into K')
//   online softmax, O = softmax(scores) @ V
// Output stored TRANSPOSED: hoT[b, e=h*64+d, s]  — makes the reference's
// transpose(1,2)+view scramble a contiguous reshape for the final GEMM.
// ---------------------------------------------------------------------------
__global__ void __launch_bounds__(128)
attn_kernel(const float* __restrict__ q, const float* __restrict__ kp,
            const float* __restrict__ v, float* __restrict__ hoT)
{
  __shared__ float p_lds[4][16][18];   // per-wave P-tile relayout scratch
  const int bid  = blockIdx.x;         // b*512 + h*32 + sblk
  const int sblk = bid & 31;
  const int h    = (bid >> 5) & 15;
  const int b    = bid >> 9;
  const int lane = threadIdx.x & 31;
  const int wave = threadIdx.x >> 5;
  const int g    = lane >> 4;
  const int ln   = lane & 15;

  const size_t bh    = (size_t)(b * HEADS + h) * SEQ * DHEAD;
  const int    qbase = sblk * 64 + wave * 16;

  // Resident Q fragments (16 rows x 64 d).
  v2f qfrag[16];
  {
    const float* qrow = q + bh + (size_t)(qbase + ln) * DHEAD;
#pragma unroll
    for (int kb = 0; kb < 16; ++kb) {
      const int c0 = kb * 4 + 2 * g;
      qfrag[kb].x = qrow[c0];
      qfrag[kb].y = qrow[c0 + 1];
    }
  }

  v8f oacc[4];
#pragma unroll
  for (int nt = 0; nt < 4; ++nt)
#pragma unroll
    for (int r = 0; r < 8; ++r) oacc[nt][r] = 0.0f;

  float m[8], l[8];
#pragma unroll
  for (int r = 0; r < 8; ++r) { m[r] = -1e30f; l[r] = 0.0f; }

  const int jb_end = qbase >> 4;       // causal: key blocks 0..qbase/16
  for (int jb = 0; jb <= jb_end; ++jb) {
    // ---- S tile = Q(16x64) @ K'^T(64x16) ----
    v8f st;
#pragma unroll
    for (int r = 0; r < 8; ++r) st[r] = 0.0f;
    {
      const float* krow = kp + bh + (size_t)(jb * 16 + ln) * DHEAD;
#pragma unroll
      for (int kb = 0; kb < 16; ++kb) {
        const int c0 = kb * 4 + 2 * g;
        v2f bf;
        bf.x = krow[c0];
        bf.y = krow[c0 + 1];
        st = wmma_f32(qfrag[kb], bf, st);
      }
    }

    // ---- causal mask + online softmax update ----
    const int j = jb * 16 + ln;
    float pr[8], alpha[8];
#pragma unroll
    for (int r = 0; r < 8; ++r) {
      const int i = qbase + r + 8 * g;
      float sv = st[r];
      if (j > i) sv = -2e8f;            // reference mask value
      const float mn = fmaxf(m[r], hmax16(sv));
      const float a  = __expf(m[r] - mn);
      const float p  = __expf(sv - mn);
      l[r]     = l[r] * a + hsum16(p);
      m[r]     = mn;
      alpha[r] = a;
      pr[r]    = p;
    }
#pragma unroll
    for (int nt = 0; nt < 4; ++nt)
#pragma unroll
      for (int r = 0; r < 8; ++r) oacc[nt][r] *= alpha[r];

    // ---- relayout P: C-layout -> A-layout via per-wave LDS ----
#pragma unroll
    for (int r = 0; r < 8; ++r) p_lds[wave][r + 8 * g][ln] = pr[r];
    __builtin_amdgcn_wave_barrier();    // keep LDS store->load ordered (intra-wave, HW in-order)
    v2f pfrag[4];
#pragma unroll
    for (int kc = 0; kc < 4; ++kc) {
      pfrag[kc].x = p_lds[wave][ln][kc * 4 + 2 * g];
      pfrag[kc].y = p_lds[wave][ln][kc * 4 + 2 * g + 1];
    }
    __builtin_amdgcn_wave_barrier();

    // ---- O += P(16x16) @ V(16x64) ----
#pragma unroll
    for (int nt = 0; nt < 4; ++nt) {
      const int d = nt * 16 + ln;
#pragma unroll
      for (int kc = 0; kc < 4; ++kc) {
        const int jj = jb * 16 + kc * 4 + 2 * g;
        v2f vf;
        vf.x = v[bh + (size_t)jj * DHEAD + d];
        vf.y = v[bh + (size_t)(jj + 1) * DHEAD + d];
        oacc[nt] = wmma_f32(pfrag[kc], vf, oacc[nt]);
      }
    }
  }

  // ---- finalize, store transposed hoT[b, e, s] ----
#pragma unroll
  for (int nt = 0; nt < 4; ++nt) {
    const int e = h * DHEAD + nt * 16 + ln;
#pragma unroll
    for (int r = 0; r < 8; ++r) {
      const int s = qbase + r + 8 * g;
      hoT[((size_t)b * EMB + e) * SEQ + s] = oacc[nt][r] / l[r];
    }
  }
}

// ---------------------------------------------------------------------------
// Kernel 3: out[b,i,o] = sum_j Y[b,i,j] * Wc[o,j] + bc[o]
// where Y[b] == reshape(hoT[b], (2048,1024))  (the reference scramble, free).
// ---------------------------------------------------------------------------
__global__ void __launch_bounds__(128)
proj_out_kernel(const float* __restrict__ hoT, const float* __restrict__ Wc,
                const float* __restrict__ bc, float* __restrict__ out)
{
  const int bid  = blockIdx.x;         // b*512 + iblk*16 + oblk
  const int oblk = bid & 15;
  const int iblk = (bid >> 4) & 31;
  const int b    = bid >> 9;
  const int lane = threadIdx.x & 31;
  const int wave = threadIdx.x >> 5;
  const int g    = lane >> 4;
  const int ln   = lane & 15;
  const int ibase = iblk * 64 + wave * 16;
  const int obase = oblk * 64;

  const float* Y    = hoT + (size_t)b * SEQ * EMB;
  const float* yrow = Y + (size_t)(ibase + ln) * EMB;

  v8f acc[4];
#pragma unroll
  for (int nt = 0; nt < 4; ++nt)
#pragma unroll
    for (int r = 0; r < 8; ++r) acc[nt][r] = 0.0f;

  for (int kb = 0; kb < EMB / 4; ++kb) {
    const int c0 = kb * 4 + 2 * g;
    v2f af;
    af.x = yrow[c0];
    af.y = yrow[c0 + 1];
#pragma unroll
    for (int nt = 0; nt < 4; ++nt) {
      const int o = obase + nt * 16 + ln;
      v2f bf;
      bf.x = Wc[(size_t)o * EMB + c0];
      bf.y = Wc[(size_t)o * EMB + c0 + 1];
      acc[nt] = wmma_f32(af, bf, acc[nt]);
    }
  }
#pragma unroll
  for (int nt = 0; nt < 4; ++nt) {
    const int o = obase + nt * 16 + ln;
    const float bias = bc[o];
#pragma unroll
    for (int r = 0; r < 8; ++r) {
      const int i = ibase + r + 8 * g;
      out[((size_t)b * SEQ + i) * EMB + o] = acc[nt][r] + bias;
    }
  }
}

extern "C" void kernel_launch(void* const* d_in, const int* in_sizes, int n_in,
                              void* d_out, int out_size, void* d_ws, size_t ws_size,
                              hipStream_t stream) {
  (void)in_sizes; (void)n_in; (void)out_size; (void)ws_size;
  const float* x       = (const float*)d_in[0];
  const float* Wq      = (const float*)d_in[1];
  const float* Wk      = (const float*)d_in[2];
  const float* Wv      = (const float*)d_in[3];
  const float* rel_pos = (const float*)d_in[4];
  const float* Wc      = (const float*)d_in[5];
  const float* bc      = (const float*)d_in[6];
  float* out = (float*)d_out;

  const size_t N = (size_t)BATCH * HEADS * SEQ * DHEAD;  // 4M floats each
  float* q   = (float*)d_ws;
  float* kp  = q + N;
  float* v   = kp + N;
  float* hoT = v + N;

  qkv_kernel<<<dim3(BATCH * HEADS * (SEQ / 64), 3), 128, 0, stream>>>(
      x, Wq, Wk, Wv, rel_pos, q, kp, v);
  attn_kernel<<<dim3(BATCH * HEADS * (SEQ / 64)), 128, 0, stream>>>(q, kp, v, hoT);
  proj_out_kernel<<<dim3(BATCH * (SEQ / 64) * (EMB / 64)), 128, 0, stream>>>(
      hoT, Wc, bc, out);
}